// IFHMM_62062277427634
// MI455X (gfx1250) — compile-verified
//
#include <hip/hip_runtime.h>
#include <math.h>

typedef __attribute__((ext_vector_type(2))) float v2f;
typedef __attribute__((ext_vector_type(4))) float v4f;
typedef __attribute__((ext_vector_type(8))) float v8f;

#define NS 16      // states
#define NC 8       // chains
#define NN 1500    // nodes
#define NT 500     // iters
#define TP 512     // T padded to 16
#define NP 1504    // N padded to 16

// workspace layout (float offsets)
#define OFF_LSI   0                       // (S,C)            128
#define OFF_LCW   (OFF_LSI + NS*NC)       // (T,C)            4000
#define OFF_LT    (OFF_LCW + NT*NC)       // (C,S,S)          2048
#define OFF_LE    (OFF_LT + NC*NS*NS)     // (S,N)            24000
#define OFF_ETEXP (OFF_LE + NS*NN)        // (NP,S) exp(lE)^T 24064
#define OFF_W     (OFF_ETEXP + NP*NS)     // (TP,S)           8192
#define OFF_M     (OFF_W + TP*NS)         // (TP)             512
#define OFF_ADD   (OFF_M + TP)            // (T,S,C)          64000

// ---------- stage 1: small row log-softmaxes -------------------------------
__global__ void k_norm_small(const float* __restrict__ in_si,
                             const float* __restrict__ in_cw,
                             const float* __restrict__ in_tm,
                             float* __restrict__ lsi,
                             float* __restrict__ lcw,
                             float* __restrict__ lT) {
  int j = blockIdx.x * blockDim.x + threadIdx.x;
  if (j < NC) {                       // state_init: softmax over states (stride NC)
    float mx = -INFINITY;
    for (int s = 0; s < NS; ++s) mx = fmaxf(mx, in_si[s*NC + j]);
    float sum = 0.f;
    for (int s = 0; s < NS; ++s) sum += expf(in_si[s*NC + j] - mx);
    float lse = mx + logf(sum);
    for (int s = 0; s < NS; ++s) lsi[s*NC + j] = in_si[s*NC + j] - lse;
  } else if (j < NC + NC*NS) {        // transition rows: contiguous 16
    int r = j - NC;
    const float* p = in_tm + r*NS;  float* q = lT + r*NS;
    float mx = -INFINITY;
    for (int k = 0; k < NS; ++k) mx = fmaxf(mx, p[k]);
    float sum = 0.f;
    for (int k = 0; k < NS; ++k) sum += expf(p[k] - mx);
    float lse = mx + logf(sum);
    for (int k = 0; k < NS; ++k) q[k] = p[k] - lse;
  } else if (j < NC + NC*NS + NT) {   // chain weights rows: contiguous 8
    int r = j - NC - NC*NS;
    const float* p = in_cw + r*NC;  float* q = lcw + r*NC;
    float mx = -INFINITY;
    for (int k = 0; k < NC; ++k) mx = fmaxf(mx, p[k]);
    float sum = 0.f;
    for (int k = 0; k < NC; ++k) sum += expf(p[k] - mx);
    float lse = mx + logf(sum);
    for (int k = 0; k < NC; ++k) q[k] = p[k] - lse;
  }
}

// ---------- stage 2: emission log-softmax over N, plus exp() transposed ----
__global__ void k_norm_emis(const float* __restrict__ in_E,
                            float* __restrict__ lE,
                            float* __restrict__ EtExp) {
  int s = blockIdx.x;
  __shared__ float red[256];
  const float* row = in_E + s*NN;
  float mx = -INFINITY;
  for (int n = threadIdx.x; n < NN; n += 256) mx = fmaxf(mx, row[n]);
  red[threadIdx.x] = mx; __syncthreads();
  for (int o = 128; o > 0; o >>= 1) {
    if (threadIdx.x < o) red[threadIdx.x] = fmaxf(red[threadIdx.x], red[threadIdx.x+o]);
    __syncthreads();
  }
  mx = red[0]; __syncthreads();
  float sum = 0.f;
  for (int n = threadIdx.x; n < NN; n += 256) sum += expf(row[n] - mx);
  red[threadIdx.x] = sum; __syncthreads();
  for (int o = 128; o > 0; o >>= 1) {
    if (threadIdx.x < o) red[threadIdx.x] += red[threadIdx.x+o];
    __syncthreads();
  }
  float lse = mx + logf(red[0]);
  for (int n = threadIdx.x; n < NP; n += 256) {
    if (n < NN) {
      float v = row[n] - lse;
      lE[s*NN + n] = v;
      EtExp[n*NS + s] = expf(v);
    } else {
      EtExp[n*NS + s] = 0.f;          // N-padding contributes nothing
    }
  }
}

// ---------- stage 3: sequential HMM prior scan (1 block, 128 threads) ------
__global__ void k_scan(const float* __restrict__ lsi,
                       const float* __restrict__ lT,
                       float* __restrict__ out3) {
  __shared__ float hsh[NS*NC];
  __shared__ float lTsh[NC*NS*NS];
  int tid = threadIdx.x;              // 0..127 -> (s' = tid>>3, c = tid&7)
  for (int i = tid; i < NC*NS*NS; i += 128) lTsh[i] = lT[i];
  float h = lsi[tid];
  hsh[tid] = h;
  out3[tid] = h;
  __syncthreads();
  int sp = tid >> 3, c = tid & 7;
  const float* tc = &lTsh[c*(NS*NS) + sp];   // [c][s][sp], stride NS over s
  for (int t = 1; t < NT; ++t) {
    float vals[NS];
    float mx = -INFINITY;
#pragma unroll
    for (int s = 0; s < NS; ++s) {
      float v = hsh[s*NC + c] + tc[s*NS];
      vals[s] = v;
      mx = fmaxf(mx, v);
    }
    float sum = 0.f;
#pragma unroll
    for (int s = 0; s < NS; ++s) sum += expf(vals[s] - mx);
    float nv = mx + logf(sum);
    __syncthreads();
    hsh[tid] = nv;
    out3[t*NS*NC + tid] = nv;
    __syncthreads();
  }
}

// ---------- stage 4: per-t prep: addval, m[t], W[t][s] ---------------------
__global__ void k_prep(const float* __restrict__ lh,   // = out3
                       const float* __restrict__ lcw,
                       float* __restrict__ addval,
                       float* __restrict__ Wm,
                       float* __restrict__ mArr) {
  int t = blockIdx.x;
  int tid = threadIdx.x;              // 128
  __shared__ float red[128];
  if (t >= NT) {                      // zero-pad rows so WMMA A-tiles are clean
    if (tid < NS) Wm[t*NS + tid] = 0.f;
    if (tid == 0) mArr[t] = 0.f;
    return;
  }
  int s = tid >> 3, c = tid & 7;
  float a = lh[t*(NS*NC) + tid] + lcw[t*NC + c];
  addval[t*(NS*NC) + tid] = a;
  red[tid] = a; __syncthreads();
  for (int o = 64; o > 0; o >>= 1) {
    if (tid < o) red[tid] = fmaxf(red[tid], red[tid+o]);
    __syncthreads();
  }
  float m = red[0]; __syncthreads();
  red[tid] = expf(a - m); __syncthreads();
  if (c == 0) {
    float sum = 0.f;
    for (int k = 0; k < NC; ++k) sum += red[s*NC + k];
    Wm[t*NS + s] = sum;
  }
  if (tid == 0) mArr[t] = m;
}

// ---------- stage 5: out1 = m[t] + log(W @ E) via V_WMMA_F32_16X16X4_F32 ---
// One wave per 16x16 tile; K=16 = 4 chained WMMAs with K=4.
// A (16x4 f32): lane l (0..15) holds A[M=l][K=kb..kb+1], lanes 16..31 K=kb+2..kb+3.
// B (4x16 f32): lane l holds B[K][N=l] with same half/K striping -> load from E^T.
__global__ void k_wmma_out1(const float* __restrict__ Wm,
                            const float* __restrict__ mArr,
                            const float* __restrict__ EtExp,
                            float* __restrict__ out1) {
  int tileN = blockIdx.x, tileT = blockIdx.y;
  int lane = threadIdx.x;
  int half = lane >> 4, l = lane & 15;
  int t0 = tileT * 16, n0 = tileN * 16;
  const float* arow = Wm    + (t0 + l) * NS;
  const float* brow = EtExp + (n0 + l) * NS;
  v8f acc = {};
#pragma unroll
  for (int kc = 0; kc < 4; ++kc) {
    int kb = kc*4 + half*2;
    v2f a = *(const v2f*)(arow + kb);
    v2f b = *(const v2f*)(brow + kb);
    acc = __builtin_amdgcn_wmma_f32_16x16x4_f32(
        /*neg_a=*/false, a, /*neg_b=*/false, b,
        /*c_mod=*/(short)0, acc, /*reuse_a=*/false, /*reuse_b=*/false);
  }
  int nn = n0 + l;
#pragma unroll
  for (int r = 0; r < 8; ++r) {
    int tt = t0 + half*8 + r;       // D layout: VGPR r -> M=r (lanes 0-15) / r+8 (16-31)
    if (tt < NT && nn < NN) out1[tt*NN + nn] = mArr[tt] + logf(acc[r]);
  }
}

// ---------- stage 6: out2[t,s,c,n] = lE[s,n] + addval[t,s,c]  (384 MB) -----
__global__ void k_bcast_out2(const float* __restrict__ lE,
                             const float* __restrict__ addval,
                             float* __restrict__ out2) {
  int row = blockIdx.x;               // (t*16+s)*8+c
  int s = (row >> 3) & 15;
  float add = addval[row];
  const v4f* ef = (const v4f*)(lE + s*NN);
  v4f* of = (v4f*)(out2 + (size_t)row * NN);
  for (int i = threadIdx.x; i < NN/4; i += blockDim.x) {
    v4f v = ef[i];
    v += add;
    __builtin_nontemporal_store(v, of + i);   // output >> L2; bypass-rinse
  }
}

extern "C" void kernel_launch(void* const* d_in, const int* in_sizes, int n_in,
                              void* d_out, int out_size, void* d_ws, size_t ws_size,
                              hipStream_t stream) {
  const float* in_si = (const float*)d_in[0];   // (S,C)
  const float* in_cw = (const float*)d_in[1];   // (T,C)
  const float* in_E  = (const float*)d_in[2];   // (S,1,N)
  const float* in_tm = (const float*)d_in[3];   // (C,S,S)

  float* out  = (float*)d_out;
  float* out1 = out;                               // (T,N)
  float* out2 = out1 + (size_t)NT*NN;              // (T,S,C,N)
  float* out3 = out2 + (size_t)NT*NS*NC*NN;        // (T,S,C)

  float* ws     = (float*)d_ws;
  float* lsi    = ws + OFF_LSI;
  float* lcw    = ws + OFF_LCW;
  float* lT     = ws + OFF_LT;
  float* lE     = ws + OFF_LE;
  float* EtExp  = ws + OFF_ETEXP;
  float* Wm     = ws + OFF_W;
  float* mArr   = ws + OFF_M;
  float* addval = ws + OFF_ADD;

  k_norm_small<<<3, 256, 0, stream>>>(in_si, in_cw, in_tm, lsi, lcw, lT);
  k_norm_emis<<<NS, 256, 0, stream>>>(in_E, lE, EtExp);
  k_scan<<<1, 128, 0, stream>>>(lsi, lT, out3);
  k_prep<<<TP, 128, 0, stream>>>(out3, lcw, addval, Wm, mArr);
  dim3 gw(NP/16, TP/16);
  k_wmma_out1<<<gw, 32, 0, stream>>>(Wm, mArr, EtExp, out1);
  k_bcast_out2<<<NT*NS*NC, 128, 0, stream>>>(lE, addval, out2);
}